// Self_Attention_header_29222957482445
// MI455X (gfx1250) — compile-verified
//
#include <hip/hip_runtime.h>

typedef __attribute__((ext_vector_type(2))) float v2f;
typedef __attribute__((ext_vector_type(8))) float v8f;

#define S_LEN 16384
#define EMB 16
#define WAVES_PER_BLOCK 8
#define BLOCK_THREADS (WAVES_PER_BLOCK * 32)
#define KV_BLOCKS (S_LEN / 32)   // 2 tiles (32 kv columns) per iteration

__launch_bounds__(BLOCK_THREADS, 1)
__global__ void flash_attn_f32_wmma(
    const float* __restrict__ x,
    const float* __restrict__ Wq, const float* __restrict__ bq,
    const float* __restrict__ Wk, const float* __restrict__ bk,
    const float* __restrict__ Wv, const float* __restrict__ bv,
    float* __restrict__ out)
{
  const int lane = threadIdx.x & 31;
  const int wave = threadIdx.x >> 5;
  const int h    = lane >> 4;   // K-dim half this lane holds (A/B operand layouts)
  const int n    = lane & 15;   // A-row / B-col / C-col index

  const int q0 = (blockIdx.x * WAVES_PER_BLOCK + wave) * 16;

  // 1/sqrt(E) * log2(e): scores come out of the QK WMMA already in the
  // log2 domain, so softmax uses bare v_exp_f32 (exp2) with no extra muls.
  const float scale = 0.25f * 1.44269504088896340736f;

  // K-dim element map for the 4 chained 16x16x4 WMMA steps:
  // step t = i>>1 covers k = 4t..4t+3; this lane holds k = 4t + 2h + (i&1).
  int emap[8];
#pragma unroll
  for (int i = 0; i < 8; ++i) emap[i] = 4 * (i >> 1) + 2 * h + (i & 1);

  // Per-lane constant weights (16 floats each; WGP$/L2 resident).
  float wqs[8], bqs[8], wk8[8], bk8[8];
#pragma unroll
  for (int i = 0; i < 8; ++i) {
    wqs[i] = Wq[emap[i]] * scale;
    bqs[i] = bq[emap[i]] * scale;
    wk8[i] = Wk[emap[i]];
    bk8[i] = bk[emap[i]];
  }
  const float wvn = Wv[n];
  const float bvn = bv[n];

  // Q^T tile as the B operand of S^T = K @ Q^T:
  // B-layout: lane holds column m = n, K-elems e = emap[i].  Scale folded in.
  const float xq = x[q0 + n];
  float qa[8];
#pragma unroll
  for (int i = 0; i < 8; ++i) qa[i] = xq * wqs[i] + bqs[i];

  // O^T accumulator (C/D layout): acc[r] = O[row n][e = r + 8h].
  // Online-softmax stats for query row n are per-lane scalars (log2 domain).
  v8f acc = {0.f, 0.f, 0.f, 0.f, 0.f, 0.f, 0.f, 0.f};
  float mrun = -__builtin_inff();
  float lrun = 0.f;

  for (int j = 0; j < KV_BLOCKS; ++j) {
    const int kv0 = j * 32;
    const float xk0 = x[kv0 + n];
    const float xk1 = x[kv0 + 16 + n];
    const int pf = (kv0 + 32) & (S_LEN - 1);           // wrap on last iter
    __builtin_prefetch(&x[pf + n], 0, 3);

    // K tiles as A operands: A(row = kv-local n, k = e = emap[i]).
    float kb0[8], kb1[8];
#pragma unroll
    for (int i = 0; i < 8; ++i) {
      kb0[i] = xk0 * wk8[i] + bk8[i];
      kb1[i] = xk1 * wk8[i] + bk8[i];
    }

    // V^T tiles as A operands: A(row = e = n, k = c = emap[i]) = V[kv0+c][n].
    // x rows come in adjacent pairs -> b64 loads (uniform per half).
    float vb0[8], vb1[8];
#pragma unroll
    for (int t = 0; t < 4; ++t) {
      const float2 xv0 = *reinterpret_cast<const float2*>(x + kv0 + 4 * t + 2 * h);
      const float2 xv1 = *reinterpret_cast<const float2*>(x + kv0 + 16 + 4 * t + 2 * h);
      vb0[2 * t]     = xv0.x * wvn + bvn;
      vb0[2 * t + 1] = xv0.y * wvn + bvn;
      vb1[2 * t]     = xv1.x * wvn + bvn;
      vb1[2 * t + 1] = xv1.y * wvn + bvn;
    }

    // S^T = K @ (Q*scale)^T for both tiles: 2 x 4 chained V_WMMA_F32_16X16X4_F32.
    // C layout: stX[r] = log2-domain score(query n, kv column r + 8h of tile X).
    v8f st0 = {0.f, 0.f, 0.f, 0.f, 0.f, 0.f, 0.f, 0.f};
    v8f st1 = {0.f, 0.f, 0.f, 0.f, 0.f, 0.f, 0.f, 0.f};
#pragma unroll
    for (int t = 0; t < 4; ++t) {
      v2f b = {qa[2 * t], qa[2 * t + 1]};
      v2f a0 = {kb0[2 * t], kb0[2 * t + 1]};
      v2f a1 = {kb1[2 * t], kb1[2 * t + 1]};
      st0 = __builtin_amdgcn_wmma_f32_16x16x4_f32(false, a0, false, b, (short)0, st0,
                                                  false, false);
      st1 = __builtin_amdgcn_wmma_f32_16x16x4_f32(false, a1, false, b, (short)0, st1,
                                                  false, false);
    }

    // Online softmax over 32 columns: 16 in-lane + 16 in the other half.
    float mx = fmaxf(st0[0], st1[0]);
#pragma unroll
    for (int r = 1; r < 8; ++r) mx = fmaxf(mx, fmaxf(st0[r], st1[r]));
    mx = fmaxf(mx, __shfl_xor(mx, 16, 32));
    const float mnew = fmaxf(mrun, mx);

    float p0[8], p1[8];
    float ps = 0.f;
#pragma unroll
    for (int r = 0; r < 8; ++r) {
      p0[r] = __builtin_amdgcn_exp2f(st0[r] - mnew);
      p1[r] = __builtin_amdgcn_exp2f(st1[r] - mnew);
      ps += p0[r] + p1[r];
    }
    ps += __shfl_xor(ps, 16, 32);

    const float corr = __builtin_amdgcn_exp2f(mrun - mnew);
    lrun = lrun * corr + ps;
    mrun = mnew;
#pragma unroll
    for (int r = 0; r < 8; ++r) acc[r] *= corr;

    // P^T: C layout (comp r -> kv row r+8h) -> B layout (comp i -> kv row emap[i]).
    // 4 of 8 elements per tile cross the half boundary: xor-16 shuffles.
    float pb0[8], pb1[8];
    {
      const float r0 = __shfl_xor(h ? p0[0] : p0[2], 16, 32);
      const float r1 = __shfl_xor(h ? p0[1] : p0[3], 16, 32);
      const float r2 = __shfl_xor(h ? p0[4] : p0[6], 16, 32);
      const float r3 = __shfl_xor(h ? p0[5] : p0[7], 16, 32);
      pb0[0] = h ? r0 : p0[0];
      pb0[1] = h ? r1 : p0[1];
      pb0[2] = h ? r2 : p0[4];
      pb0[3] = h ? r3 : p0[5];
      pb0[4] = h ? p0[2] : r0;
      pb0[5] = h ? p0[3] : r1;
      pb0[6] = h ? p0[6] : r2;
      pb0[7] = h ? p0[7] : r3;
    }
    {
      const float r0 = __shfl_xor(h ? p1[0] : p1[2], 16, 32);
      const float r1 = __shfl_xor(h ? p1[1] : p1[3], 16, 32);
      const float r2 = __shfl_xor(h ? p1[4] : p1[6], 16, 32);
      const float r3 = __shfl_xor(h ? p1[5] : p1[7], 16, 32);
      pb1[0] = h ? r0 : p1[0];
      pb1[1] = h ? r1 : p1[1];
      pb1[2] = h ? r2 : p1[4];
      pb1[3] = h ? r3 : p1[5];
      pb1[4] = h ? p1[2] : r0;
      pb1[5] = h ? p1[3] : r1;
      pb1[6] = h ? p1[6] : r2;
      pb1[7] = h ? p1[7] : r3;
    }

    // O^T += V^T @ P^T for both tiles: 2 x 4 chained V_WMMA_F32_16X16X4_F32.
#pragma unroll
    for (int t = 0; t < 4; ++t) {
      v2f a0 = {vb0[2 * t], vb0[2 * t + 1]};
      v2f b0 = {pb0[2 * t], pb0[2 * t + 1]};
      acc = __builtin_amdgcn_wmma_f32_16x16x4_f32(false, a0, false, b0, (short)0, acc,
                                                  false, false);
    }
#pragma unroll
    for (int t = 0; t < 4; ++t) {
      v2f a1 = {vb1[2 * t], vb1[2 * t + 1]};
      v2f b1 = {pb1[2 * t], pb1[2 * t + 1]};
      acc = __builtin_amdgcn_wmma_f32_16x16x4_f32(false, a1, false, b1, (short)0, acc,
                                                  false, false);
    }
  }

  // Normalize (stats are in-lane) and store: 8 consecutive floats per lane
  // -> two coalesced b128 stores; halves cover e = 0..7 / 8..15 of row n.
  const float inv_l = 1.0f / lrun;
  float* base = out + (q0 + n) * EMB + 8 * h;
  float4 o0, o1;
  o0.x = acc[0] * inv_l; o0.y = acc[1] * inv_l;
  o0.z = acc[2] * inv_l; o0.w = acc[3] * inv_l;
  o1.x = acc[4] * inv_l; o1.y = acc[5] * inv_l;
  o1.z = acc[6] * inv_l; o1.w = acc[7] * inv_l;
  *reinterpret_cast<float4*>(base)     = o0;
  *reinterpret_cast<float4*>(base + 4) = o1;
}

extern "C" void kernel_launch(void* const* d_in, const int* in_sizes, int n_in,
                              void* d_out, int out_size, void* d_ws, size_t ws_size,
                              hipStream_t stream) {
  (void)in_sizes; (void)n_in; (void)out_size; (void)d_ws; (void)ws_size;
  const float* x  = (const float*)d_in[0];
  const float* Wq = (const float*)d_in[1];
  const float* bq = (const float*)d_in[2];
  const float* Wk = (const float*)d_in[3];
  const float* bk = (const float*)d_in[4];
  const float* Wv = (const float*)d_in[5];
  const float* bv = (const float*)d_in[6];
  float* out = (float*)d_out;

  dim3 grid(S_LEN / (16 * WAVES_PER_BLOCK));   // 128 blocks x 8 waves = 1024 query tiles
  dim3 block(BLOCK_THREADS);
  hipLaunchKernelGGL(flash_attn_f32_wmma, grid, block, 0, stream,
                     x, Wq, bq, Wk, bk, Wv, bv, out);
}